// Betti3DLoss_81544249082441
// MI455X (gfx1250) — compile-verified
//
#include <hip/hip_runtime.h>
#include <stdint.h>

// p_hat: (B=8, C=4, D=160, H=160, W=64) fp32, contiguous.
// Element strides:
//   b: 4*160*160*64 = 6,553,600
//   c: 160*160*64   = 1,638,400
//   d: 160*64       = 10,240
//   h: 64
// Needed samples (exact resize): d = 5i+2, h = 5j+2, w in {8k+3, 8k+4}.
// Per block (one batch b, one slice s of 8 i-values): 4-D TDM tile
//   dim0 (x):  2 contiguous floats (the w pair)
//   dim1 (y):  k = 0..7,  stride  8 elements
//   dim2 (z):  j = 0..31, stride  320 elements (5*64)
//   dim3 (zz): i-sub = 0..7, stride 51200 elements (5*160*64)
// => 2*8*32*8 = 4096 floats = 16 KB into LDS per block.

typedef unsigned int u32x4 __attribute__((ext_vector_type(4)));
typedef int          i32x8 __attribute__((ext_vector_type(8)));
typedef int          i32x4 __attribute__((ext_vector_type(4)));

#define STRIDE_B 6553600ull
#define STRIDE_C 1638400ull
#define STRIDE_D 10240ull
#define NPAIRS   2048      // pairs per block = 8*32*8
#define TILE_F   4096      // floats per block tile

__global__ __launch_bounds__(256)
void betti_tdm_tile_max(const float* __restrict__ p,
                        const int* __restrict__ sid_p,
                        float* __restrict__ ws)
{
    __shared__ float tile[TILE_F];
    __shared__ float wred[8];

    const unsigned b = blockIdx.x;   // 0..7 batch
    const unsigned s = blockIdx.y;   // 0..3 i-slice (8 i each)
    const int sid    = *sid_p;       // channel select (uniform scalar load)

    // Global byte address of tile start: (b, sid, d=40s+2, h=2, w=3)
    const unsigned long long elem_off =
        (unsigned long long)b * STRIDE_B +
        (unsigned long long)sid * STRIDE_C +
        (unsigned long long)(40u * s + 2u) * STRIDE_D +
        2ull * 64ull + 3ull;
    const unsigned long long gaddr =
        (unsigned long long)(uintptr_t)p + 4ull * elem_off;
    const unsigned lds_addr = (unsigned)(uintptr_t)(&tile[0]); // addr[31:0] = LDS offset

    if (threadIdx.x == 0) {
        // ---- D# group 0: count=1 | lds_addr | global_addr[56:0] | type=2 ----
        u32x4 g0;
        g0.x = 1u;                                   // count=1, is_restore=0, gather off
        g0.y = lds_addr;                             // bits 63:32
        g0.z = (unsigned)(gaddr & 0xffffffffull);    // bits 95:64
        g0.w = (unsigned)((gaddr >> 32) & 0x01ffffffull) | (2u << 30); // addr[56:32] | type=2

        // ---- D# group 1 ----
        i32x8 g1;
        g1[0] = 0x00020000;           // workgroup_mask=0, data_size=2 (4B)
        g1[1] = (int)(2u << 16);      // [31:16]=tensor_dim0[15:0]=2
        g1[2] = (int)(8u << 16);      // [15:0]=tensor_dim0[31:16]=0, [31:16]=tensor_dim1[15:0]=8
        g1[3] = (int)(2u << 16);      // [15:0]=tensor_dim1[31:16]=0, [31:16]=tile_dim0=2
        g1[4] = (int)(8u | (32u << 16)); // tile_dim1=8, tile_dim2=32
        g1[5] = 8;                    // tensor_dim0_stride[31:0] = 8
        g1[6] = (int)(320u << 16);    // stride0[47:32]=0, tensor_dim1_stride[15:0]=320
        g1[7] = 0;                    // tensor_dim1_stride[47:16]=0

        // ---- D# group 2 ----
        i32x4 g2;
        g2[0] = 32;                   // tensor_dim2 = 32
        g2[1] = 8;                    // tensor_dim3 = 8
        g2[2] = 51200;                // tensor_dim2_stride[31:0] = 5*160*64
        g2[3] = (int)(8u << 16);      // stride2[47:32]=0, tile_dim3 = 8

        // ---- D# group 3 (unused 5th dim) ----
        i32x4 g3 = {0, 0, 0, 0};

        // clang-23 / therock-10.0 form: 6 args, extra i32x8 operand (zeros), cpol last
        i32x8 gz = {0, 0, 0, 0, 0, 0, 0, 0};
        __builtin_amdgcn_tensor_load_to_lds(g0, g1, g2, g3, gz, /*cpol=*/0);
        __builtin_amdgcn_s_wait_tensorcnt(0);
    }
    __syncthreads();

    // Pair-average + per-thread max over 8 pairs, values in [0,1)
    float m = -1.0f;
    #pragma unroll
    for (int r = 0; r < 8; ++r) {
        const int q = (int)threadIdx.x + r * 256;   // pair index, q < NPAIRS
        const float a = tile[2 * q];
        const float c = tile[2 * q + 1];
        m = fmaxf(m, 0.5f * (a + c));
    }

    // wave32 reduction
    #pragma unroll
    for (int off = 16; off > 0; off >>= 1)
        m = fmaxf(m, __shfl_down(m, off, 32));
    if ((threadIdx.x & 31u) == 0u)
        wred[threadIdx.x >> 5] = m;
    __syncthreads();

    if (threadIdx.x == 0) {
        float mm = wred[0];
        #pragma unroll
        for (int i = 1; i < 8; ++i) mm = fmaxf(mm, wred[i]);
        ws[b * 4u + s] = mm;   // block max for (batch b, slice s)
    }
}

__global__ void betti_finalize(const float* __restrict__ ws,
                               const int* __restrict__ sid_p,
                               float* __restrict__ out)
{
    if (threadIdx.x == 0 && blockIdx.x == 0) {
        const int sid = *sid_p;
        const int T[4][3] = {{1,0,0},{1,0,0},{1,1,0},{1,0,0}};
        const int F[3]    = {1,0,0};
        int be = 0;
        #pragma unroll
        for (int k = 0; k < 3; ++k) {
            int d = F[k] - T[sid][k];
            be += (d < 0) ? -d : d;
        }
        float total = 0.0f;
        if (be > 0) {
            for (int b = 0; b < 8; ++b) {
                float mm = ws[b * 4];
                for (int s = 1; s < 4; ++s) mm = fmaxf(mm, ws[b * 4 + s]);
                total += (1.0f - mm) * (float)be;
            }
        }
        out[0] = total / 8.0f;
    }
}

extern "C" void kernel_launch(void* const* d_in, const int* in_sizes, int n_in,
                              void* d_out, int out_size, void* d_ws, size_t ws_size,
                              hipStream_t stream)
{
    (void)in_sizes; (void)n_in; (void)out_size; (void)ws_size;
    const float* p_hat = (const float*)d_in[0];
    const int*   sid   = (const int*)d_in[1];
    float*       out   = (float*)d_out;
    float*       ws    = (float*)d_ws;   // 32 floats: [batch*4 + slice]

    dim3 grid(8, 4, 1);
    betti_tdm_tile_max<<<grid, 256, 0, stream>>>(p_hat, sid, ws);
    betti_finalize<<<1, 32, 0, stream>>>(ws, sid, out);
}